// QuantumCircuitLayer_88828513616277
// MI455X (gfx1250) — compile-verified
//
#include <hip/hip_runtime.h>
#include <hip/hip_bf16.h>
#include <math.h>

typedef __attribute__((ext_vector_type(2))) float v2f;
typedef __attribute__((ext_vector_type(8))) float v8f;

// G = [Re(U) (16x16) ; Im(U) (16x16)], row = output amp j, col = input amp k.
__device__ float g_G[512];

// ---------------------------------------------------------------------------
// Prep: simulate the fixed variational unitary U on the 16 basis states.
// Lane k owns basis state |k>; after the gate sequence its amplitudes are
// column k of U. Wire 0 is the most-significant bit (PennyLane convention):
// mask(wire i) = 8 >> i. Runs once on one wave -> keep accurate sincosf.
// ---------------------------------------------------------------------------
__global__ void qc_prep(const float* __restrict__ W) {
    const int k = threadIdx.x;
    if (k >= 16) return;
    float re[16], im[16];
#pragma unroll
    for (int j = 0; j < 16; ++j) { re[j] = (j == k) ? 1.0f : 0.0f; im[j] = 0.0f; }

#pragma unroll
    for (int d = 0; d < 2; ++d) {
        // CNOT ring: (0->1), (1->2), (2->3), (3->0)
        const int ctl[4] = {0, 1, 2, 3};
        const int tgt[4] = {1, 2, 3, 0};
#pragma unroll
        for (int g = 0; g < 4; ++g) {
            const int mc = 8 >> ctl[g], mt = 8 >> tgt[g];
            float tr[16], ti[16];
#pragma unroll
            for (int j = 0; j < 16; ++j) {
                const int src = (j & mc) ? (j ^ mt) : j;
                tr[j] = re[src]; ti[j] = im[src];
            }
#pragma unroll
            for (int j = 0; j < 16; ++j) { re[j] = tr[j]; im[j] = ti[j]; }
        }
        // Per-wire RY(w0) then RZ(w1)
#pragma unroll
        for (int i = 0; i < 4; ++i) {
            const int mi = 8 >> i;
            float sy, cy; sincosf(W[d * 8 + i * 2 + 0] * 0.5f, &sy, &cy);
#pragma unroll
            for (int j = 0; j < 16; ++j) {
                if (!(j & mi)) {
                    const int jb = j | mi;
                    const float ar = re[j], ai = im[j], br = re[jb], bi = im[jb];
                    re[j]  = cy * ar - sy * br;  im[j]  = cy * ai - sy * bi;
                    re[jb] = sy * ar + cy * br;  im[jb] = sy * ai + cy * bi;
                }
            }
            float sz, cz; sincosf(W[d * 8 + i * 2 + 1] * 0.5f, &sz, &cz);
#pragma unroll
            for (int j = 0; j < 16; ++j) {
                const float r = re[j], q = im[j];
                if (j & mi) { re[j] = r * cz - q * sz; im[j] = q * cz + r * sz; } // * e^{+i phi/2}
                else        { re[j] = r * cz + q * sz; im[j] = q * cz - r * sz; } // * e^{-i phi/2}
            }
        }
    }
#pragma unroll
    for (int j = 0; j < 16; ++j) {
        g_G[j * 16 + k]       = re[j];
        g_G[256 + j * 16 + k] = im[j];
    }
}

// ---------------------------------------------------------------------------
// Main: 16 samples per wave. Lane pair (n, n+16) owns sample n of the tile;
// low lanes build amplitudes with b2=0, high lanes b2=1 (matches the f32
// 16x16x4 WMMA B-operand layout: K split low/high half-wave).
// Trig uses the hardware trans unit (v_sin/v_cos) -> co-executes with VALU.
// ---------------------------------------------------------------------------
__global__ __launch_bounds__(256) void qc_main(const float4* __restrict__ x4,
                                               float4* __restrict__ out4,
                                               int B) {
    const int lane = threadIdx.x & 31;
    const int wave = threadIdx.x >> 5;
    const int base = (blockIdx.x * 8 + wave) * 16;   // 16 samples per wave
    if (base + 16 > B) return;                        // wave-uniform guard
    const int hi = lane >> 4;
    const int s  = base + (lane & 15);

    // Embedding angles: RY(x*pi) -> uses cos/sin of x*pi/2
    const float HP = 1.57079632679489662f;
    const float4 xs = x4[s];
    float c0, s0, c1, s1, c2, s2, c3, s3;
    __sincosf(xs.x * HP, &s0, &c0);
    __sincosf(xs.y * HP, &s1, &c1);
    __sincosf(xs.z * HP, &s2, &c2);
    __sincosf(xs.w * HP, &s3, &c3);
    const float f2 = hi ? s2 : c2;   // this half-wave's b2 factor

    // psi0[j] = f0(b0)*f1(b1)*f2*f3(b3),  j = 8b0+4b1+2b2+b3, chunk q = 2b0+b1
    const float u0 = c0 * c1 * f2;
    const float u1 = c0 * s1 * f2;
    const float u2 = s0 * c1 * f2;
    const float u3 = s0 * s1 * f2;
    v2f bv[4];
    bv[0] = (v2f){u0 * c3, u0 * s3};
    bv[1] = (v2f){u1 * c3, u1 * s3};
    bv[2] = (v2f){u2 * c3, u2 * s3};
    bv[3] = (v2f){u3 * c3, u3 * s3};

    // A operand: lane m holds G[row=m][k0 + 2*hi], G[row=m][k0 + 2*hi + 1]
    const float* Grow = g_G + (lane & 15) * 16 + 2 * hi;

    v8f dr = {0.f, 0.f, 0.f, 0.f, 0.f, 0.f, 0.f, 0.f};
    v8f di = dr;
#pragma unroll
    for (int q = 0; q < 4; ++q) {
        const v2f ar = *(const v2f*)(Grow + 4 * q);
        const v2f ai = *(const v2f*)(Grow + 256 + 4 * q);
        dr = __builtin_amdgcn_wmma_f32_16x16x4_f32(false, ar, false, bv[q],
                                                   (short)0, dr, false, false);
        di = __builtin_amdgcn_wmma_f32_16x16x4_f32(false, ai, false, bv[q],
                                                   (short)0, di, false, false);
    }

    // Lane holds y_{j} for j = r + 8*hi (r = 0..7) of its sample.
    float p[8];
#pragma unroll
    for (int r = 0; r < 8; ++r) p[r] = dr[r] * dr[r] + di[r] * di[r];

    // <Z_w> partial sums with sign = (1 - 2*bit_w(j)), bit_w = (j >> (3-w)) & 1
    const float a = p[0] + p[1], b = p[2] + p[3], c = p[4] + p[5], d = p[6] + p[7];
    const float ab = a + b, cd = c + d;
    const float sumAll = ab + cd;
    float e0 = hi ? -sumAll : sumAll;                       // bit0 = hi
    float e1 = ab - cd;                                     // bit1 = r>>2
    float e2 = (a - b) + (c - d);                           // bit2 = (r>>1)&1
    float e3 = (p[0] - p[1]) + (p[2] - p[3]) + (p[4] - p[5]) + (p[6] - p[7]);

    e0 += __shfl_xor(e0, 16, 32);
    e1 += __shfl_xor(e1, 16, 32);
    e2 += __shfl_xor(e2, 16, 32);
    e3 += __shfl_xor(e3, 16, 32);

    if (lane < 16) out4[s] = make_float4(e0, e1, e2, e3);
}

extern "C" void kernel_launch(void* const* d_in, const int* in_sizes, int n_in,
                              void* d_out, int out_size, void* d_ws, size_t ws_size,
                              hipStream_t stream) {
    const float*  x  = (const float*)d_in[0];   // [B,4] float32
    const float*  w  = (const float*)d_in[1];   // [2,4,2] float32
    float*        out = (float*)d_out;          // [B,4] float32
    (void)d_ws; (void)ws_size; (void)n_in; (void)out_size;

    const int B = in_sizes[0] / 4;

    qc_prep<<<1, 32, 0, stream>>>(w);

    const int tiles  = B / 16;                  // B = 1<<20 -> exact
    const int blocks = (tiles + 7) / 8;         // 8 waves (128 samples) per block
    qc_main<<<blocks, 256, 0, stream>>>((const float4*)x, (float4*)out, B);
}